// Test_method_78623671320838
// MI455X (gfx1250) — compile-verified
//
#include <hip/hip_runtime.h>
#include <hip/hip_bf16.h>

// Problem constants (match reference)
constexpr int Bn = 50000;
constexpr int Wn = 100;
constexpr int Ln = 10;
constexpr int Hn = 10;
constexpr int Nn = Wn * Ln;     // 1000 : GEMM N dim, n = m*W + w  (== flat [L,W] layout)
constexpr int NP = 1008;        // N padded to 63 tiles of 16
constexpr int KP = 12;          // K = L padded to 3 WMMA k-steps of 4
constexpr int TILE_B = 16;      // batch items per block (WMMA M tile)
constexpr float NEGV = -99999999.0f;

typedef __attribute__((ext_vector_type(2))) float v2f;
typedef __attribute__((ext_vector_type(8))) float v8f;

// T2B pair layout: pairRow = s*2 + half  (s = k-step, half: lanes0-15 -> K=4s,4s+1 ; lanes16-31 -> K=4s+2,4s+3)
// T2B[pairRow][col] = { T2[4s+2*half][col], T2[4s+2*half+1][col] },  col = m*W + w, 6 pair rows x NP cols.

__global__ void t2_zero_kernel(float* __restrict__ t2) {
    int i = blockIdx.x * blockDim.x + threadIdx.x;
    if (i < 6 * NP * 2) t2[i] = 0.0f;
}

__global__ void t2_fill_kernel(const float* __restrict__ wabi, float* __restrict__ t2) {
    int t = blockIdx.x * blockDim.x + threadIdx.x;
    if (t >= Wn * Ln) return;
    int w = t / Ln, l = t % Ln;               // l is the GEMM K index
    const float* row = wabi + (size_t)(w * Ln + l) * Ln;
    float v[Ln];
    float mx = -3.0e38f;
    #pragma unroll
    for (int i = 0; i < Ln; ++i) { v[i] = row[i]; mx = fmaxf(mx, v[i]); }
    float s = 0.0f;
    #pragma unroll
    for (int i = 0; i < Ln; ++i) { v[i] = __expf(v[i] - mx); s += v[i]; }
    float inv = 1.0f / s;
    int pairRow = (l / 4) * 2 + ((l % 4) / 2);  // 0..4 (row 5 = K pad, stays zero)
    int comp    = l & 1;
    #pragma unroll
    for (int m = 0; m < Ln; ++m) {
        int col = m * Wn + w;
        t2[(pairRow * NP + col) * 2 + comp] = v[m] * inv;
    }
}

// ---------------- fused main kernel: one block = 16 batch items -------------------------
__launch_bounds__(256)
__global__ void fused_crowd_kernel(const float* __restrict__ hidden,
                                   const float* __restrict__ cmask,
                                   const float* __restrict__ wemb,
                                   const float* __restrict__ t2,
                                   float* __restrict__ out_f,
                                   float* __restrict__ out_ob,
                                   float* __restrict__ out_soft,
                                   float* __restrict__ out_att) {
    // s_big: phase 1-3 = crowd_mask tile [16][1000]; phase 4-5 = ob_labels tile [16][1000]
    __shared__ float s_big[TILE_B * Nn];       // 64000 B
    __shared__ float s_att[TILE_B * Wn];       // 6400 B
    __shared__ float s_fs [TILE_B * KP];       // 768 B  (K-padded f_score)
    __shared__ float s_emb[Wn * Hn];           // 4000 B
    __shared__ float s_hid[TILE_B * Hn];       // 640 B

    const int tid = threadIdx.x;
    const int b0  = blockIdx.x * TILE_B;

    // -- stage crowd_mask tile with CDNA5 async global->LDS B128 (ASYNCcnt, no VGPR staging) --
    {
        unsigned long long gbase = (unsigned long long)(const void*)(cmask + (size_t)b0 * Nn);
        unsigned int       lbase = (unsigned int)(unsigned long long)(const void*)s_big; // addr[31:0] = LDS offset
        for (int i = tid; i < TILE_B * Nn / 4; i += 256) {   // 4000 x 16B transfers
            unsigned long long ga = gbase + (unsigned long long)i * 16u;
            unsigned int       la = lbase + (unsigned int)i * 16u;
            asm volatile("global_load_async_to_lds_b128 %0, %1, off"
                         :: "v"(la), "v"(ga) : "memory");
        }
    }
    // overlap: small constant tables via normal loads
    for (int i = tid; i < Wn * Hn; i += 256) s_emb[i] = wemb[i];
    for (int i = tid; i < TILE_B * Hn; i += 256) s_hid[i] = hidden[(size_t)b0 * Hn + i];
    asm volatile("s_wait_asynccnt 0" ::: "memory");
    __syncthreads();

    // -- phase 1: raw att[it][w] = dot(h,e) * rowsum(cmask); 0 -> NEG sentinel --
    for (int p = tid; p < TILE_B * Wn; p += 256) {
        int it = p / Wn, w = p % Wn;
        float d = 0.0f;
        #pragma unroll
        for (int k = 0; k < Hn; ++k) d += s_hid[it * Hn + k] * s_emb[w * Hn + k];
        float cs = 0.0f;
        #pragma unroll
        for (int l = 0; l < Ln; ++l) cs += s_big[it * Nn + w * Ln + l];
        float a = d * cs;
        s_att[p] = (a == 0.0f) ? NEGV : a;
    }
    __syncthreads();

    // -- phase 2: softmax over W per item; 16 lanes (half-wave) per item, xor-shuffle reduce --
    {
        int g = tid >> 4;          // item 0..15
        int j = tid & 15;
        float mx = -3.0e38f;
        for (int w = j; w < Wn; w += 16) mx = fmaxf(mx, s_att[g * Wn + w]);
        #pragma unroll
        for (int off = 8; off; off >>= 1) mx = fmaxf(mx, __shfl_xor(mx, off, 16));
        float s = 0.0f;
        for (int w = j; w < Wn; w += 16) {
            float e = __expf(s_att[g * Wn + w] - mx);   // exp(NEG-mx) underflows to 0 exactly
            s_att[g * Wn + w] = e;
            s += e;
        }
        #pragma unroll
        for (int off = 8; off; off >>= 1) s += __shfl_xor(s, off, 16);
        float inv = 1.0f / s;
        for (int w = j; w < Wn; w += 16) {
            float v = s_att[g * Wn + w] * inv;
            s_att[g * Wn + w] = v;
            out_att[(size_t)(b0 + g) * Wn + w] = v;
        }
    }
    __syncthreads();

    // -- phase 3: f_score[it][l] = sum_w att * cmask ; zero-pad K rows 10,11 --
    if (tid < TILE_B * Ln) {
        int it = tid / Ln, l = tid % Ln;
        float acc = 0.0f;
        for (int w = 0; w < Wn; ++w) acc += s_att[it * Wn + w] * s_big[it * Nn + w * Ln + l];
        s_fs[it * KP + l] = acc;
        out_f[(size_t)(b0 + it) * Ln + l] = acc;
    } else if (tid < TILE_B * Ln + 32) {
        int idx = tid - TILE_B * Ln;                 // 0..31
        s_fs[(idx >> 1) * KP + Ln + (idx & 1)] = 0.0f;
    }
    __syncthreads();

    // -- phase 4: ob_labels = f_score[16 x 12] @ T2[12 x 1008] via V_WMMA_F32_16X16X4_F32 --
    // ISA 7.12.2 fp32 A/B layout: lanes 0-15 hold K={4s,4s+1}, lanes 16-31 hold K={4s+2,4s+3}.
    {
        int wv   = tid >> 5;                 // wave 0..7 -> N tiles wv, wv+8, ...
        int lane = tid & 31;
        int hrow = lane & 15;                // A: row M ; B: col within tile
        int kb   = (lane < 16) ? 0 : 2;
        int half = (lane < 16) ? 0 : 1;
        v2f a0 = { s_fs[hrow * KP + 0 + kb], s_fs[hrow * KP + 1 + kb] };
        v2f a1 = { s_fs[hrow * KP + 4 + kb], s_fs[hrow * KP + 5 + kb] };
        v2f a2 = { s_fs[hrow * KP + 8 + kb], s_fs[hrow * KP + 9 + kb] };
        const v2f* t2b = (const v2f*)t2;
        for (int nt = wv; nt < NP / 16; nt += 8) {
            int col = nt * 16 + hrow;
            v2f bb0 = t2b[(0 + half) * NP + col];   // one b64 load per k-step
            v2f bb1 = t2b[(2 + half) * NP + col];
            v2f bb2 = t2b[(4 + half) * NP + col];
            v8f c = {};
            c = __builtin_amdgcn_wmma_f32_16x16x4_f32(false, a0, false, bb0, (short)0, c, false, false);
            c = __builtin_amdgcn_wmma_f32_16x16x4_f32(false, a1, false, bb1, (short)0, c, false, false);
            c = __builtin_amdgcn_wmma_f32_16x16x4_f32(false, a2, false, bb2, (short)0, c, false, false);
            if (col < Nn) {
                #pragma unroll
                for (int r = 0; r < 8; ++r) {
                    int row = (lane < 16) ? r : (r + 8);   // C layout: VGPR r -> M=r / M=r+8
                    s_big[row * Nn + col] = c[r];
                    out_ob[(size_t)(b0 + row) * Nn + col] = c[r];
                }
            }
        }
    }
    __syncthreads();

    // -- phase 5: ob_labels_soft = softmax over m (stride W in the tile), per (item, w) --
    for (int p = tid; p < TILE_B * Wn; p += 256) {
        int it = p / Wn, w = p % Wn;
        float v[Ln];
        float mx = -3.0e38f;
        #pragma unroll
        for (int m = 0; m < Ln; ++m) { v[m] = s_big[it * Nn + m * Wn + w]; mx = fmaxf(mx, v[m]); }
        float s = 0.0f;
        #pragma unroll
        for (int m = 0; m < Ln; ++m) { v[m] = __expf(v[m] - mx); s += v[m]; }
        float inv = 1.0f / s;
        #pragma unroll
        for (int m = 0; m < Ln; ++m)
            out_soft[(size_t)(b0 + it) * Nn + m * Wn + w] = v[m] * inv;
    }
}

extern "C" void kernel_launch(void* const* d_in, const int* in_sizes, int n_in,
                              void* d_out, int out_size, void* d_ws, size_t ws_size,
                              hipStream_t stream) {
    const float* hidden = (const float*)d_in[0];   // [B,H]
    const float* cmask  = (const float*)d_in[1];   // [B,W,L]
    const float* wemb   = (const float*)d_in[2];   // [W,H]
    const float* wabi   = (const float*)d_in[3];   // [W,L,L]

    float* t2 = (float*)d_ws;                      // T2B pair-layout B matrix, 6*NP*2 floats = 48384 B

    float* out      = (float*)d_out;               // outputs concat: f_score, ob, ob_soft, att
    float* out_f    = out;
    float* out_ob   = out_f + (size_t)Bn * Ln;
    float* out_soft = out_ob + (size_t)Bn * Nn;
    float* out_att  = out_soft + (size_t)Bn * Nn;

    t2_zero_kernel<<<(6 * NP * 2 + 255) / 256, 256, 0, stream>>>(t2);
    t2_fill_kernel<<<(Wn * Ln + 255) / 256, 256, 0, stream>>>(wabi, t2);
    fused_crowd_kernel<<<Bn / TILE_B, 256, 0, stream>>>(hidden, cmask, wemb, t2,
                                                        out_f, out_ob, out_soft, out_att);
}